// LorentzBatchNorm_allvar_21199958573186
// MI455X (gfx1250) — compile-verified
//
#include <hip/hip_runtime.h>

// ---------------------------------------------------------------------------
// Lorentz BatchNorm (training path) for x:(64,4096,128) f32 on gfx1250.
// Bandwidth-bound: ~536MB total traffic; x (134MB) fits in the 192MB L2, so
// the 2nd/3rd passes over x should hit L2. Output stores are non-temporal to
// keep x resident. Pass-1 S-reduction runs on the matrix core via
// V_WMMA_F32_16X16X4_F32 with a ones A-matrix (layout-robust reduction).
// Bulk passes use b128 loads/stores (f = lane*4+j mapping) to minimize VMEM
// instruction count at 23.3 TB/s.
// ---------------------------------------------------------------------------

typedef float v2f __attribute__((ext_vector_type(2)));
typedef float v4f __attribute__((ext_vector_type(4)));
typedef float v8f __attribute__((ext_vector_type(8)));

#define EPS_F    1e-5f
#define CLAMP_F  1e-8f
#define MAXE_F   32.0f

#define B_DIM 64
#define S_DIM 4096
#define D_DIM 128
#define NROWS (B_DIM * S_DIM)

__device__ __forceinline__ float wave_sum32(float v) {
    v += __shfl_xor(v, 16, 32);
    v += __shfl_xor(v, 8, 32);
    v += __shfl_xor(v, 4, 32);
    v += __shfl_xor(v, 2, 32);
    v += __shfl_xor(v, 1, 32);
    return v;
}

// ---------------------------------------------------------------------------
// Kernel 1: per-(batch, S-chunk) column sums via WMMA f32 16x16x4.
// grid = 64*16 blocks, block = 256 (8 waves). Wave w covers 16 columns.
// A = ones(16x4) -> D[m,n] = sum over 4 rows of B column n (all m identical).
// ---------------------------------------------------------------------------
__global__ __launch_bounds__(256) void k1_colsum_wmma(const float* __restrict__ x,
                                                      float* __restrict__ pp) {
    const int tid   = threadIdx.x;
    const int wid   = tid >> 5;
    const int lane  = tid & 31;
    const int half  = lane >> 4;     // 0 or 1
    const int col16 = lane & 15;
    const int b     = blockIdx.x >> 4;
    const int chunk = blockIdx.x & 15;
    const int col   = wid * 16 + col16;

    const long base = ((long)b * S_DIM + (long)chunk * 256) * D_DIM;
    const float* xp = x + base + col;

    v2f a; a.x = 1.0f; a.y = 1.0f;       // ones A-matrix (every slot = 1)
    v8f c0 = {}; v8f c1 = {};

    for (int r0 = 0; r0 < 256; r0 += 8) {
        v2f b0, b1;
        b0.x = xp[(long)(r0 + 0 + half) * D_DIM];
        b0.y = xp[(long)(r0 + 2 + half) * D_DIM];
        b1.x = xp[(long)(r0 + 4 + half) * D_DIM];
        b1.y = xp[(long)(r0 + 6 + half) * D_DIM];
        c0 = __builtin_amdgcn_wmma_f32_16x16x4_f32(false, a, false, b0,
                                                   (short)0, c0, false, false);
        c1 = __builtin_amdgcn_wmma_f32_16x16x4_f32(false, a, false, b1,
                                                   (short)0, c1, false, false);
    }
    if (lane < 16) pp[(long)blockIdx.x * D_DIM + col] = c0[0] + c1[0];
}

// ---------------------------------------------------------------------------
// Kernel 2: double Lorentz centroid -> mean vector (D,). 1 block, 128 threads.
// ---------------------------------------------------------------------------
__global__ __launch_bounds__(128) void k2_mean(const float* __restrict__ pp,
                                               float* __restrict__ gmean) {
    const int f    = threadIdx.x;    // feature 0..127
    const int lane = f & 31;
    const int wid  = f >> 5;
    __shared__ float red[4];
    __shared__ float sh0;

    float acc = 0.0f;
    for (int b = 0; b < B_DIM; ++b) {
        float s = 0.0f;
        for (int c = 0; c < 16; ++c)
            s += pp[((long)(b * 16 + c)) * D_DIM + f];
        float avg = s * (1.0f / (float)S_DIM);

        float ws = wave_sum32(avg * avg);
        if (lane == 0) red[wid] = ws;
        if (f == 0)    sh0 = avg;
        __syncthreads();
        float tot = red[0] + red[1] + red[2] + red[3];
        float a0  = sh0;
        __syncthreads();
        float negl = 2.0f * a0 * a0 - tot;             // -<avg,avg>_L
        acc += avg / sqrtf(fmaxf(negl, CLAMP_F));
    }

    float avg2 = acc * (1.0f / (float)B_DIM);
    float ws2  = wave_sum32(avg2 * avg2);
    if (lane == 0) red[wid] = ws2;
    if (f == 0)    sh0 = avg2;
    __syncthreads();
    float tot2  = red[0] + red[1] + red[2] + red[3];
    float a02   = sh0;
    float negl2 = 2.0f * a02 * a02 - tot2;
    gmean[f] = avg2 / sqrtf(fmaxf(negl2, CLAMP_F));
}

// ---------------------------------------------------------------------------
// Shared per-row tangent computation: logmap at mean + transport to origin.
// One wave per row; lane handles features {4*lane .. 4*lane+3} -> a single
// fully-coalesced b128 load per row. Feature 0 is lane 0, element 0.
// ---------------------------------------------------------------------------
__device__ __forceinline__ void row_tangent(const float* __restrict__ xp,
                                            const v4f mv, float m0,
                                            int lane, float v[4]) {
    v4f xv = ((const v4f*)xp)[lane];           // global_load_b128

    float dot = xv.x*mv.x + xv.y*mv.y + xv.z*mv.z + xv.w*mv.w;
    dot = wave_sum32(dot);
    float x0    = __shfl(xv.x, 0, 32);
    float alpha = fmaxf(2.0f * m0 * x0 - dot, 1.0f + 1e-7f);  // -<mean,x>_L

    float u[4];
#pragma unroll
    for (int j = 0; j < 4; ++j) u[j] = xv[j] - alpha * mv[j];
    float usq = u[0]*u[0] + u[1]*u[1] + u[2]*u[2] + u[3]*u[3];
    usq = wave_sum32(usq);
    float u0    = x0 - alpha * m0;
    float unorm = sqrtf(fmaxf(usq - 2.0f * u0 * u0, CLAMP_F));
    float fac   = acoshf(alpha) / unorm;

    // parallel transport mean -> origin: v += (-v0/(1+m0)) * (mean + o)
    float coef = -(fac * u0) / (1.0f + m0);
#pragma unroll
    for (int j = 0; j < 4; ++j) {
        float vv = fac * u[j] + coef * mv[j];
        if (j == 0 && lane == 0) vv += coef;   // origin time component
        v[j] = vv;
    }
}

// ---------------------------------------------------------------------------
// Kernel 3: per-feature sum / sumsq of x_T over all rows (deterministic
// per-block partials, no atomics). grid = 512, block = 256 (8 waves).
// ---------------------------------------------------------------------------
__global__ __launch_bounds__(256) void k3_stats(const float* __restrict__ x,
                                                const float* __restrict__ gmean,
                                                float* __restrict__ part) {
    const int tid  = threadIdx.x;
    const int wid  = tid >> 5;
    const int lane = tid & 31;

    const v4f mv = ((const v4f*)gmean)[lane];
    const float m0 = gmean[0];

    float sA[4] = {0, 0, 0, 0}, qA[4] = {0, 0, 0, 0};
    for (int r = blockIdx.x * 8 + wid; r < NROWS; r += gridDim.x * 8) {
        float v[4];
        row_tangent(x + (long)r * D_DIM, mv, m0, lane, v);
#pragma unroll
        for (int j = 0; j < 4; ++j) { sA[j] += v[j]; qA[j] += v[j] * v[j]; }
    }

    __shared__ float ls[8][D_DIM];
    __shared__ float lq[8][D_DIM];
#pragma unroll
    for (int j = 0; j < 4; ++j) {
        ls[wid][lane * 4 + j] = sA[j];
        lq[wid][lane * 4 + j] = qA[j];
    }
    __syncthreads();
    if (tid < D_DIM) {
        float s = 0.0f, q = 0.0f;
#pragma unroll
        for (int w = 0; w < 8; ++w) { s += ls[w][tid]; q += lq[w][tid]; }
        part[(long)blockIdx.x * 256 + tid]          = s;
        part[(long)blockIdx.x * 256 + D_DIM + tid]  = q;
    }
}

// ---------------------------------------------------------------------------
// Kernel 4: reduce partials -> inv_std[f] = gamma[f]/(sqrt(var+eps)+eps).
// ---------------------------------------------------------------------------
__global__ __launch_bounds__(128) void k4_finalize(const float* __restrict__ part,
                                                   const float* __restrict__ gamma,
                                                   float* __restrict__ inv) {
    const int f = threadIdx.x;
    float s = 0.0f, q = 0.0f;
    for (int k = 0; k < 512; ++k) {
        s += part[(long)k * 256 + f];
        q += part[(long)k * 256 + D_DIM + f];
    }
    const float invN = 1.0f / (float)NROWS;
    float mu  = s * invN;
    float var = fmaxf(q * invN - mu * mu, 0.0f);
    float st  = sqrtf(var + EPS_F);
    inv[f] = gamma[f] / (st + EPS_F);
}

// ---------------------------------------------------------------------------
// Kernel 5: apply — recompute x_T (x should hit L2), scale, max-Euclid
// rescale, transport to beta, expmap at beta; b128 non-temporal stores.
// grid = 2048, block = 256 (8 waves).
// ---------------------------------------------------------------------------
__global__ __launch_bounds__(256) void k5_apply(const float* __restrict__ x,
                                                const float* __restrict__ gmean,
                                                const float* __restrict__ inv,
                                                const float* __restrict__ beta,
                                                float* __restrict__ out) {
    const int tid  = threadIdx.x;
    const int wid  = tid >> 5;
    const int lane = tid & 31;

    const v4f mv = ((const v4f*)gmean)[lane];
    const v4f iv = ((const v4f*)inv)[lane];
    const v4f bv = ((const v4f*)beta)[lane];
    const float m0 = gmean[0];
    const float b0 = beta[0];

    for (int r = blockIdx.x * 8 + wid; r < NROWS; r += gridDim.x * 8) {
        float t[4];
        row_tangent(x + (long)r * D_DIM, mv, m0, lane, t);
#pragma unroll
        for (int j = 0; j < 4; ++j) t[j] *= iv[j];

        // rescale_to_max_euclid
        float nsq = t[0]*t[0] + t[1]*t[1] + t[2]*t[2] + t[3]*t[3];
        nsq = wave_sum32(nsq);
        float n  = sqrtf(nsq);
        float sc = fminf(1.0f, MAXE_F / fmaxf(n, CLAMP_F));
#pragma unroll
        for (int j = 0; j < 4; ++j) t[j] *= sc;

        // transport origin -> beta
        float dbt = bv.x*t[0] + bv.y*t[1] + bv.z*t[2] + bv.w*t[3];
        dbt = wave_sum32(dbt);
        float t0    = __shfl(t[0], 0, 32);
        float coef2 = (dbt - 2.0f * b0 * t0) / (1.0f + b0);   // <beta,t>_L/(1+b0)
        float w[4];
#pragma unroll
        for (int j = 0; j < 4; ++j) {
            float wv = t[j] + coef2 * bv[j];
            if (j == 0 && lane == 0) wv += coef2;             // + coef*o
            w[j] = wv;
        }
        float w0  = t0 + coef2 * (b0 + 1.0f);
        float wsq = w[0]*w[0] + w[1]*w[1] + w[2]*w[2] + w[3]*w[3];
        wsq = wave_sum32(wsq);
        float nu  = sqrtf(fmaxf(wsq - 2.0f * w0 * w0, CLAMP_F));
        float ch  = coshf(nu);
        float shn = sinhf(nu) / nu;

        v4f res;
#pragma unroll
        for (int j = 0; j < 4; ++j) res[j] = ch * bv[j] + shn * w[j];
        __builtin_nontemporal_store(res, (v4f*)(out + (long)r * D_DIM) + lane);
    }
}

// ---------------------------------------------------------------------------
extern "C" void kernel_launch(void* const* d_in, const int* in_sizes, int n_in,
                              void* d_out, int out_size, void* d_ws, size_t ws_size,
                              hipStream_t stream) {
    const float* x     = (const float*)d_in[0];
    const float* gamma = (const float*)d_in[1];
    const float* beta  = (const float*)d_in[2];
    float*       out   = (float*)d_out;

    float* ws    = (float*)d_ws;
    float* pp    = ws;                 // 1024*128 floats, reused as 512*256 stats partials
    float* gmean = ws + 131072;        // 128
    float* inv   = ws + 131072 + 128;  // 128

    k1_colsum_wmma<<<B_DIM * 16, 256, 0, stream>>>(x, pp);
    k2_mean<<<1, 128, 0, stream>>>(pp, gmean);
    k3_stats<<<512, 256, 0, stream>>>(x, gmean, pp);      // reuse pp region
    k4_finalize<<<1, 128, 0, stream>>>(pp, gamma, inv);
    k5_apply<<<2048, 256, 0, stream>>>(x, gmean, inv, beta, out);
}